// Tweak_26139170963708
// MI455X (gfx1250) — compile-verified
//
#include <hip/hip_runtime.h>

#define Bdim 64
#define Udim 512
#define Cdim 512
#define Rdim 512
#define NSTEPS 10

typedef float v2f __attribute__((ext_vector_type(2)));
typedef float v8f __attribute__((ext_vector_type(8)));

// D = A(16x4) * B(4x16) + C, fp32 WMMA (CDNA5 V_WMMA_F32_16X16X4_F32)
__device__ __forceinline__ v8f wmma4(v2f a, v2f b, v8f c) {
  return __builtin_amdgcn_wmma_f32_16x16x4_f32(false, a, false, b, (short)0, c,
                                               false, false);
}

__device__ __forceinline__ float sigm(float x) {
  return 1.0f / (1.0f + __expf(-x));
}

// Accumulate xrow[k] * wrow[k] over K into the 16x16 WMMA accumulator.
// Per ISA layout: lane l holds A[row=l&15][kh], A[row][kh+1] with kh=(l>>4)*2,
// and B[kh][col=l&15], B[kh+1][col] (B = W^T tile, i.e. W row (n) contiguous).
__device__ __forceinline__ void mm_acc(const float* __restrict__ xrow,
                                       const float* __restrict__ wrow, int K,
                                       v8f& acc, int kh) {
#pragma unroll 8
  for (int k = 0; k < K; k += 4) {
    v2f a = *(const v2f*)(xrow + k + kh);
    v2f b = *(const v2f*)(wrow + k + kh);
    acc = wmma4(a, b, acc);
  }
}

// Fused LSTM cell: gates = [x0|x1] @ Wih^T + h @ Whh^T + bih + bhh, then
// pointwise update. 4 waves per block: wave w computes the i/f/g/o tile at
// gate column w*R + n0; tiles combined through LDS for the nonlinearity.
__global__ void __launch_bounds__(128)
k_lstm_cell(const float* __restrict__ x0, const float* __restrict__ x1,
            const float* __restrict__ h_in, const float* __restrict__ c_in,
            const float* __restrict__ Wih, int wih_stride,
            const float* __restrict__ Whh, const float* __restrict__ bih,
            const float* __restrict__ bhh, float* __restrict__ h_out,
            float* __restrict__ c_out) {
  const int wave = threadIdx.x >> 5;
  const int lane = threadIdx.x & 31;
  const int rc = lane & 15;          // A row / B col within tile
  const int kh = (lane >> 4) * 2;    // K pair selector
  const int b0 = blockIdx.x * 16;
  const int n0 = blockIdx.y * 16;
  const int ng = wave * Rdim + n0;   // gate column base in [0, 4R)

  v8f acc = {};
  const float* wih_row = Wih + (size_t)(ng + rc) * wih_stride;
  mm_acc(x0 + (size_t)(b0 + rc) * Rdim, wih_row, Rdim, acc, kh);
  if (x1)  // uniform branch (scalar pointer): EXEC stays all-ones for WMMA
    mm_acc(x1 + (size_t)(b0 + rc) * Rdim, wih_row + Rdim, Rdim, acc, kh);
  mm_acc(h_in + (size_t)(b0 + rc) * Rdim, Whh + (size_t)(ng + rc) * Rdim, Rdim,
         acc, kh);

  __shared__ float lds[4][16][16];
  const float bias = bih[ng + rc] + bhh[ng + rc];
  const int hi = (lane >> 4) * 8;  // C/D: VGPR i -> rows i (lo half) / i+8 (hi)
#pragma unroll
  for (int i = 0; i < 8; ++i) lds[wave][i + hi][rc] = acc[i] + bias;
  __syncthreads();

  for (int e = threadIdx.x; e < 256; e += 128) {
    int r = e >> 4, cc = e & 15;
    float ig = lds[0][r][cc], fg = lds[1][r][cc];
    float gg = lds[2][r][cc], og = lds[3][r][cc];
    size_t idx = (size_t)(b0 + r) * Rdim + n0 + cc;
    float c2 = sigm(fg) * c_in[idx] + sigm(ig) * tanhf(gg);
    h_out[idx] = sigm(og) * tanhf(c2);
    c_out[idx] = c2;
  }
}

// read = relu([h|us|cs] @ Wc^T + bc), one 16x16 WMMA tile per wave.
__global__ void __launch_bounds__(128)
k_compose(const float* __restrict__ h, const float* __restrict__ us,
          const float* __restrict__ cs, const float* __restrict__ Wc,
          const float* __restrict__ bc, float* __restrict__ out) {
  const int wave = threadIdx.x >> 5;
  const int lane = threadIdx.x & 31;
  const int rc = lane & 15;
  const int kh = (lane >> 4) * 2;
  const int b0 = blockIdx.x * 16;
  const int n0 = blockIdx.y * 64 + wave * 16;

  v8f acc = {};
  const float* wrow = Wc + (size_t)(n0 + rc) * (3 * Rdim);
  mm_acc(h + (size_t)(b0 + rc) * Rdim, wrow, Rdim, acc, kh);
  mm_acc(us + (size_t)(b0 + rc) * Rdim, wrow + Rdim, Rdim, acc, kh);
  mm_acc(cs + (size_t)(b0 + rc) * Rdim, wrow + 2 * Rdim, Rdim, acc, kh);

  const float bias = bc[n0 + rc];
  const int hi = (lane >> 4) * 8;
#pragma unroll
  for (int i = 0; i < 8; ++i) {
    float v = acc[i] + bias;
    out[(size_t)(b0 + i + hi) * Rdim + n0 + rc] = fmaxf(v, 0.0f);
  }
}

// loc[b,l] = dot(Mrows[b,l,:], s[b,:]) with mask -> -inf. One wave per row.
__global__ void __launch_bounds__(256)
k_rowdot(const float* __restrict__ Mrows, const float* __restrict__ s,
         const unsigned char* __restrict__ mask, float* __restrict__ loc) {
  const int wave = threadIdx.x >> 5;
  const int lane = threadIdx.x & 31;
  const int rowIdx = blockIdx.x * 8 + wave;
  const int b = rowIdx >> 9;  // L == 512
  const float* mrow = Mrows + (size_t)rowIdx * Rdim;
  const float* sv = s + (size_t)b * Rdim;
  const int base = lane * 16;
  float acc = 0.0f;
#pragma unroll
  for (int j = 0; j < 4; ++j) {
    float4 mv = *(const float4*)(mrow + base + j * 4);
    float4 vv = *(const float4*)(sv + base + j * 4);
    acc += mv.x * vv.x + mv.y * vv.y + mv.z * vv.z + mv.w * vv.w;
  }
#pragma unroll
  for (int off = 16; off >= 1; off >>= 1) acc += __shfl_xor(acc, off, 32);
  if (lane == 0) loc[rowIdx] = mask[rowIdx] ? -__builtin_inff() : acc;
}

// Softmax over L=512 per batch; optionally z = where(mask,0,sigmoid(p)).
// Writes p both to scratch and directly into d_out (utt/cont locs).
__global__ void __launch_bounds__(512)
k_softmax(const float* __restrict__ loc, const unsigned char* __restrict__ mask,
          float* __restrict__ p, float* __restrict__ out_loc,
          float* __restrict__ z) {
  __shared__ float red[512];
  const int b = blockIdx.x, t = threadIdx.x;
  const float v = loc[b * Udim + t];
  red[t] = v;
  __syncthreads();
  for (int off = 256; off >= 1; off >>= 1) {
    if (t < off) red[t] = fmaxf(red[t], red[t + off]);
    __syncthreads();
  }
  const float m = red[0];
  __syncthreads();
  const float e = __expf(v - m);  // exp(-inf) == 0 at masked slots
  red[t] = e;
  __syncthreads();
  for (int off = 256; off >= 1; off >>= 1) {
    if (t < off) red[t] += red[t + off];
    __syncthreads();
  }
  const float pv = e / red[0];
  p[b * Udim + t] = pv;
  out_loc[b * Udim + t] = pv;
  if (z) z[b * Udim + t] = mask[b * Udim + t] ? 0.0f : sigm(pv);
}

// out[b,r] = sum_l Mrows[b,l,r] * p[b,l]; block per batch, thread per r.
__global__ void __launch_bounds__(512)
k_wsum(const float* __restrict__ Mrows, const float* __restrict__ p,
       float* __restrict__ out) {
  __shared__ float pl[512];
  const int b = blockIdx.x, t = threadIdx.x;
  pl[t] = p[b * Udim + t];
  __syncthreads();
  const float* Mb = Mrows + (size_t)b * Udim * Rdim;
  float acc = 0.0f;
#pragma unroll 4
  for (int u = 0; u < Udim; ++u) {
    if (u + 16 < Udim)  // gfx1250 global_prefetch_b8 path
      __builtin_prefetch(Mb + (size_t)(u + 16) * Rdim + t, 0, 1);
    acc = fmaf(Mb[(size_t)u * Rdim + t], pl[u], acc);
  }
  out[b * Rdim + t] = acc;
}

// Fused: g = sigmoid(hw . M_row) (masked), M1 = (1-z)*cs + z*M,
// M = g*M + (1-g)*M1. Single read + single write of M per step.
__global__ void __launch_bounds__(256)
k_update_M(float* __restrict__ M, const float* __restrict__ hw,
           const float* __restrict__ cs, const float* __restrict__ z,
           const unsigned char* __restrict__ mask) {
  const int wave = threadIdx.x >> 5;
  const int lane = threadIdx.x & 31;
  const int rowIdx = blockIdx.x * 8 + wave;
  const int b = rowIdx >> 9;  // U == 512
  float* mrow = M + (size_t)rowIdx * Rdim;
  const float* hv = hw + (size_t)b * Rdim;
  const float* cv = cs + (size_t)b * Rdim;
  const int base = lane * 16;
  float4 mv[4];
  float acc = 0.0f;
#pragma unroll
  for (int j = 0; j < 4; ++j) {
    mv[j] = *(const float4*)(mrow + base + j * 4);
    float4 h4 = *(const float4*)(hv + base + j * 4);
    acc += mv[j].x * h4.x + mv[j].y * h4.y + mv[j].z * h4.z + mv[j].w * h4.w;
  }
#pragma unroll
  for (int off = 16; off >= 1; off >>= 1) acc += __shfl_xor(acc, off, 32);
  const float g = mask[rowIdx] ? 0.0f : sigm(acc);
  const float zz = z[rowIdx];
  const float og = 1.0f - g;
#pragma unroll
  for (int j = 0; j < 4; ++j) {
    float4 c4 = *(const float4*)(cv + base + j * 4);
    float4 o;
    o.x = mv[j].x * g + ((1.0f - zz) * c4.x + mv[j].x * zz) * og;
    o.y = mv[j].y * g + ((1.0f - zz) * c4.y + mv[j].y * zz) * og;
    o.z = mv[j].z * g + ((1.0f - zz) * c4.z + mv[j].z * zz) * og;
    o.w = mv[j].w * g + ((1.0f - zz) * c4.w + mv[j].w * zz) * og;
    *(float4*)(mrow + base + j * 4) = o;
  }
}

// out[u,b,:] = M[b,u,:]
__global__ void __launch_bounds__(128)
k_transpose(const float* __restrict__ M, float* __restrict__ out) {
  const int u = blockIdx.x, b = blockIdx.y, t = threadIdx.x;
  const float4* src = (const float4*)(M + ((size_t)b * Udim + u) * Rdim);
  float4* dst = (float4*)(out + ((size_t)u * Bdim + b) * Rdim);
  dst[t] = src[t];
}

extern "C" void kernel_launch(void* const* d_in, const int* in_sizes, int n_in,
                              void* d_out, int out_size, void* d_ws,
                              size_t ws_size, hipStream_t stream) {
  const float* utt = (const float*)d_in[0];
  const float* utt_h = (const float*)d_in[1];
  const float* utt_c = (const float*)d_in[2];
  const float* cont = (const float*)d_in[3];
  const float* cont_h = (const float*)d_in[4];
  const float* cont_c = (const float*)d_in[5];
  const unsigned char* u_mask = (const unsigned char*)d_in[6];
  const unsigned char* c_mask = (const unsigned char*)d_in[7];
  const float* Wih_r = (const float*)d_in[8];
  const float* Whh_r = (const float*)d_in[9];
  const float* bih_r = (const float*)d_in[10];
  const float* bhh_r = (const float*)d_in[11];
  const float* Wc = (const float*)d_in[12];
  const float* bc = (const float*)d_in[13];
  const float* Wih_w = (const float*)d_in[14];
  const float* Whh_w = (const float*)d_in[15];
  const float* bih_w = (const float*)d_in[16];
  const float* bhh_w = (const float*)d_in[17];

  const size_t BR = (size_t)Bdim * Rdim;
  const size_t BU = (size_t)Bdim * Udim;
  const size_t MSZ = (size_t)Bdim * Udim * Rdim;
  const size_t need = MSZ + 13 * BR + 3 * BU;
  if (ws_size < need * sizeof(float)) return;

  float* W = (float*)d_ws;
  size_t off = 0;
  auto carve = [&](size_t n) { float* p = W + off; off += n; return p; };
  float* M = carve(MSZ);
  float* hr_h[2] = {carve(BR), carve(BR)};
  float* hr_c[2] = {carve(BR), carve(BR)};
  float* hw_h[2] = {carve(BR), carve(BR)};
  float* hw_c[2] = {carve(BR), carve(BR)};
  float* us[2] = {carve(BR), carve(BR)};
  float* cs[2] = {carve(BR), carve(BR)};
  float* readb = carve(BR);
  float* loc = carve(BU);
  float* pbuf = carve(BU);
  float* zbuf = carve(BU);

  // carry0 = (utt_h, utt_c, cont_h, cont_c, utt, utt_h, cont_h)
  hipMemcpyAsync(M, utt, MSZ * sizeof(float), hipMemcpyDeviceToDevice, stream);
  hipMemcpyAsync(hr_h[0], utt_h, BR * sizeof(float), hipMemcpyDeviceToDevice, stream);
  hipMemcpyAsync(hr_c[0], utt_c, BR * sizeof(float), hipMemcpyDeviceToDevice, stream);
  hipMemcpyAsync(hw_h[0], cont_h, BR * sizeof(float), hipMemcpyDeviceToDevice, stream);
  hipMemcpyAsync(hw_c[0], cont_c, BR * sizeof(float), hipMemcpyDeviceToDevice, stream);
  hipMemcpyAsync(us[0], utt_h, BR * sizeof(float), hipMemcpyDeviceToDevice, stream);
  hipMemcpyAsync(cs[0], cont_h, BR * sizeof(float), hipMemcpyDeviceToDevice, stream);

  float* out = (float*)d_out;
  float* out_M = out;
  float* out_ul = out + (size_t)Udim * Bdim * Rdim;
  float* out_cl = out_ul + (size_t)NSTEPS * Bdim * Udim;

  int cur = 0;
  for (int step = 0; step < NSTEPS; ++step) {
    const int nxt = cur ^ 1;
    // ---- Read LSTM: x = [cs|us] ----
    k_lstm_cell<<<dim3(Bdim / 16, Rdim / 16), 128, 0, stream>>>(
        cs[cur], us[cur], hr_h[cur], hr_c[cur], Wih_r, 2 * Rdim, Whh_r, bih_r,
        bhh_r, hr_h[nxt], hr_c[nxt]);
    // ---- attend over M with hr_h ----
    k_rowdot<<<dim3(Bdim * Udim / 8), 256, 0, stream>>>(M, hr_h[nxt], u_mask, loc);
    k_softmax<<<Bdim, 512, 0, stream>>>(loc, u_mask, pbuf,
                                        out_ul + (size_t)step * BU, zbuf);
    k_wsum<<<Bdim, 512, 0, stream>>>(M, pbuf, us[nxt]);
    // ---- attend over cont with us ----
    k_rowdot<<<dim3(Bdim * Cdim / 8), 256, 0, stream>>>(cont, us[nxt], c_mask, loc);
    k_softmax<<<Bdim, 512, 0, stream>>>(loc, c_mask, pbuf,
                                        out_cl + (size_t)step * BU, nullptr);
    k_wsum<<<Bdim, 512, 0, stream>>>(cont, pbuf, cs[nxt]);
    // ---- compose ----
    k_compose<<<dim3(Bdim / 16, Rdim / 64), 128, 0, stream>>>(
        hr_h[nxt], us[nxt], cs[nxt], Wc, bc, readb);
    // ---- Write LSTM ----
    k_lstm_cell<<<dim3(Bdim / 16, Rdim / 16), 128, 0, stream>>>(
        readb, nullptr, hw_h[cur], hw_c[cur], Wih_w, Rdim, Whh_w, bih_w, bhh_w,
        hw_h[nxt], hw_c[nxt]);
    // ---- fused gate-einsum + M1 + M blend (single M read/write) ----
    k_update_M<<<dim3(Bdim * Udim / 8), 256, 0, stream>>>(M, hw_h[nxt], cs[nxt],
                                                          zbuf, u_mask);
    cur = nxt;
  }
  k_transpose<<<dim3(Udim, Bdim), 128, 0, stream>>>(M, out_M);
}